// TernaryLoss_71511205478520
// MI455X (gfx1250) — compile-verified
//
#include <hip/hip_runtime.h>
#include <hip/hip_bf16.h>

typedef float v2f __attribute__((ext_vector_type(2)));
typedef float v8f __attribute__((ext_vector_type(8)));

#define H 1024
#define W 1024
#define TILE 32
#define HALO 3
#define LROWS (TILE + HALO)       // 35: rows gy0 .. gy0+34 (no top halo needed)
#define LCOLS (TILE + 2 * HALO)   // 38: cols gx0-3 .. gx0+34
#define ST 40                     // padded LDS row stride (floats)

// Full wave32 sum via V_WMMA_F32_16X16X4_F32 with an all-ones operand.
// Step 1: D = A(data) x B(ones) -> 16 row sums covering all 64 lane inputs.
// Step 2: 4 chained WMMAs (D vgpr pairs fed back as A, accumulating into C)
//         leave the total wave sum in every element of T.
// Must be called with EXEC all-1s (unconditionally, full block).
__device__ __forceinline__ float wave_reduce_wmma(float p0, float p1) {
    v2f ones; ones[0] = 1.0f; ones[1] = 1.0f;
    v2f a;    a[0] = p0;      a[1] = p1;
    v8f d = {0.f, 0.f, 0.f, 0.f, 0.f, 0.f, 0.f, 0.f};
    d = __builtin_amdgcn_wmma_f32_16x16x4_f32(false, a, false, ones, (short)0, d, false, false);
    v8f t = {0.f, 0.f, 0.f, 0.f, 0.f, 0.f, 0.f, 0.f};
    v2f a2;
    a2[0] = d[0]; a2[1] = d[1];
    t = __builtin_amdgcn_wmma_f32_16x16x4_f32(false, a2, false, ones, (short)0, t, false, false);
    a2[0] = d[2]; a2[1] = d[3];
    t = __builtin_amdgcn_wmma_f32_16x16x4_f32(false, a2, false, ones, (short)0, t, false, false);
    a2[0] = d[4]; a2[1] = d[5];
    t = __builtin_amdgcn_wmma_f32_16x16x4_f32(false, a2, false, ones, (short)0, t, false, false);
    a2[0] = d[6]; a2[1] = d[7];
    t = __builtin_amdgcn_wmma_f32_16x16x4_f32(false, a2, false, ones, (short)0, t, false, false);
    return t[0];
}

__global__ __launch_bounds__(256)
void census_loss_main(const float* __restrict__ x, const float* __restrict__ y,
                      float* __restrict__ ws) {
    __shared__ float sx[LROWS * ST];
    __shared__ float sy[LROWS * ST];
    __shared__ float wsum[8];

    const int tid = threadIdx.x;
    const int gx0 = blockIdx.x * TILE;
    const int gy0 = blockIdx.y * TILE;
    const size_t base = (size_t)blockIdx.z * (size_t)(H * W);

    // Cooperative zero-padded load: rows gy0..gy0+34, cols gx0-3..gx0+34.
    for (int i = tid; i < LROWS * LCOLS; i += 256) {
        const int r = i / LCOLS;
        const int c = i - r * LCOLS;
        const int gr = gy0 + r;
        const int gc = gx0 + c - HALO;
        float vx = 0.f, vy = 0.f;
        if (gr < H && gc >= 0 && gc < W) {
            const size_t idx = base + (size_t)gr * W + gc;
            vx = x[idx];
            vy = y[idx];
        }
        sx[r * ST + c] = vx;
        sy[r * ST + c] = vy;
    }
    __syncthreads();

    const int cx = tid & 31;   // column within tile
    const int tq = tid >> 5;   // row group 0..7
    float acc = 0.f;

    for (int r4 = 0; r4 < 4; ++r4) {
        const int lr = tq + 8 * r4;          // local row 0..31 (== LDS row)
        const int r  = gy0 + lr;             // global row
        const int c  = gx0 + cx;             // global col
        const int ci = lr * ST + (cx + HALO);
        const float xc = sx[ci];
        const float yc = sy[ci];

        const float fvp = (r >= 1 && r <= H - 2 && c >= 1 && c <= W - 2) ? 1.f : 0.f;
        float rk[4], ck[7];
#pragma unroll
        for (int k = 0; k < 4; ++k) {
            const int rr = r + k;
            rk[k] = (rr >= 1 && rr <= H - 2) ? 1.f : 0.f;
        }
#pragma unroll
        for (int j = 0; j < 7; ++j) {
            const int cc = c + j - HALO;
            ck[j] = (cc >= 1 && cc <= W - 2) ? 1.f : 0.f;
        }

        // Half the offsets: dy=0,dx=1..3  and  dy=1..3,dx=-3..3  (24 taps).
        // Each pair {p, p+o} counted once with weight valid(p)+valid(q).
        auto tap = [&](int dy, int dx) {
            const int idx = (lr + dy) * ST + (cx + HALO + dx);
            const float u = sx[idx] - xc;
            const float v = sy[idx] - yc;
            const float a = u * __builtin_amdgcn_rsqf(__builtin_fmaf(u, u, 0.81f));
            const float b = v * __builtin_amdgcn_rsqf(__builtin_fmaf(v, v, 0.81f));
            const float wv = a - b;
            const float d2 = wv * wv;
            const float T  = d2 * __builtin_amdgcn_rcpf(d2 + 0.1f);
            const float wt = __builtin_fmaf(rk[dy], ck[dx + HALO], fvp);
            acc = __builtin_fmaf(T, wt, acc);
        };
#pragma unroll
        for (int dx = 1; dx <= 3; ++dx) tap(0, dx);
#pragma unroll
        for (int dy = 1; dy <= 3; ++dy) {
#pragma unroll
            for (int dx = -3; dx <= 3; ++dx) tap(dy, dx);
        }

        // Reverse taps whose neighbor is out of image all equal
        // t0 = g(-x[p], -y[p]); closed-form count (top/left/right edges only).
        const int e0   = (HALO - c > 0) ? (HALO - c) : 0;
        const int e1   = (c - (W - 4) > 0) ? (c - (W - 4)) : 0;
        const int kout = (HALO - r > 0) ? (HALO - r) : 0;
        const int nout = e0 + 7 * kout + (3 - kout) * (e0 + e1);
        if (nout > 0 && fvp > 0.f) {
            const float a = xc * __builtin_amdgcn_rsqf(__builtin_fmaf(xc, xc, 0.81f));
            const float b = yc * __builtin_amdgcn_rsqf(__builtin_fmaf(yc, yc, 0.81f));
            const float wv = a - b;
            const float d2 = wv * wv;
            const float t0 = d2 * __builtin_amdgcn_rcpf(d2 + 0.1f);
            acc = __builtin_fmaf((float)nout, t0, acc);
        }
    }

    // Wave32 sum with v_wmma_f32_16x16x4_f32 (all lanes active here).
    const float wtot = wave_reduce_wmma(acc, 0.f);
    if ((tid & 31) == 0) wsum[tid >> 5] = wtot;
    __syncthreads();
    if (tid == 0) {
        float s = 0.f;
#pragma unroll
        for (int wv = 0; wv < 8; ++wv) s += wsum[wv];
        ws[((size_t)blockIdx.z * gridDim.y + blockIdx.y) * gridDim.x + blockIdx.x] = s;
    }
}

__global__ __launch_bounds__(256)
void census_loss_reduce(const float* __restrict__ ws, float* __restrict__ out,
                        int n, float scale) {
    __shared__ float wsum[8];
    const int tid = threadIdx.x;
    float s = 0.f;
    for (int i = tid; i < n; i += 256) s += ws[i];
    const float wtot = wave_reduce_wmma(s, 0.f);
    if ((tid & 31) == 0) wsum[tid >> 5] = wtot;
    __syncthreads();
    if (tid == 0) {
        float t = 0.f;
#pragma unroll
        for (int wv = 0; wv < 8; ++wv) t += wsum[wv];
        out[0] = t * scale;
    }
}

extern "C" void kernel_launch(void* const* d_in, const int* in_sizes, int n_in,
                              void* d_out, int out_size, void* d_ws, size_t ws_size,
                              hipStream_t stream) {
    const float* x = (const float*)d_in[0];
    const float* y = (const float*)d_in[1];
    float* out = (float*)d_out;
    float* ws  = (float*)d_ws;

    const int N = in_sizes[0] / (H * W);     // 4 images
    const int gx = W / TILE;                 // 32
    const int gy = H / TILE;                 // 32
    const int nblocks = gx * gy * N;         // 4096 partials (16 KB in d_ws)

    dim3 grid(gx, gy, N);
    census_loss_main<<<grid, 256, 0, stream>>>(x, y, ws);

    const float scale = 1.0f / (49.0f * (float)N * (float)H * (float)W);
    census_loss_reduce<<<1, 256, 0, stream>>>(ws, out, nblocks, scale);
}